// TwoAttentionGAT_39805756900145
// MI455X (gfx1250) — compile-verified
//
#include <hip/hip_runtime.h>
#include <hip/hip_bf16.h>

typedef __attribute__((ext_vector_type(2))) float v2f;
typedef __attribute__((ext_vector_type(8))) float v8f;

static constexpr int Bn = 8, Ln = 128, Dn = 512, Hn = 8, DVn = 64;
static constexpr float NEG_INF = 1000000000000.0f; // 1e12 like reference INF

__device__ __forceinline__ float leaky(float x) { return x >= 0.0f ? x : 0.2f * x; }

__device__ __forceinline__ float wave_max(float v) {
#pragma unroll
  for (int off = 16; off >= 1; off >>= 1) v = fmaxf(v, __shfl_xor(v, off, 32));
  return v;
}
__device__ __forceinline__ float wave_sum(float v) {
#pragma unroll
  for (int off = 16; off >= 1; off >>= 1) v += __shfl_xor(v, off, 32);
  return v;
}

// ---- fp32 WMMA 16x16x4 (exact fp32 path on CDNA5) -------------------------
__device__ __forceinline__ v8f wmma4(v2f a, v2f b, v8f c) {
  return __builtin_amdgcn_wmma_f32_16x16x4_f32(false, a, false, b, (short)0, c,
                                               false, false);
}

// Accumulate a 16x16 fp32 tile: C += A[row0:+16, 0:K] * B[0:K, col0:+16]
// A row-major (lda), B row-major (ldb). One wave32 per tile.
// A 16x4 layout: lane m=lane&15, VGPRs hold K=kb,kb+1 with kb=2*(lane>=16).
// B 4x16 layout: lane n=lane&15, VGPRs hold rows kb,kb+1.
__device__ __forceinline__ v8f wmma_tile(const float* __restrict__ A, int lda,
                                         const float* __restrict__ B, int ldb,
                                         int row0, int col0, int K, int lane) {
  v8f acc = {};
  const int m = lane & 15;
  const int kb = (lane >> 4) << 1;
  const float* arow = A + (size_t)(row0 + m) * lda;
  const float* bcol = B + col0 + (lane & 15);
  for (int k0 = 0; k0 < K; k0 += 4) {
    v2f a, b;
    a.x = arow[k0 + kb];
    a.y = arow[k0 + kb + 1];
    b.x = bcol[(size_t)(k0 + kb) * ldb];
    b.y = bcol[(size_t)(k0 + kb + 1) * ldb];
    acc = wmma4(a, b, acc);
  }
  return acc;
}

// ---- 0) wrc[k][h] = sum_d W_relation[k, h*64+d] * w_rel[h, d] -------------
__global__ __launch_bounds__(256) void k_wrc(const float* __restrict__ Wr,
                                             const float* __restrict__ wrel,
                                             float* __restrict__ wrc) {
  int id = blockIdx.x * blockDim.x + threadIdx.x; // 0..4095
  int k = id >> 3, h = id & 7;
  const float* wrow = Wr + (size_t)k * (Hn * DVn) + h * DVn;
  const float* vrow = wrel + h * DVn;
  float s = 0.0f;
#pragma unroll 8
  for (int d = 0; d < DVn; ++d) s = fmaf(wrow[d], vrow[d], s);
  wrc[(size_t)k * Hn + h] = s; // k-major, 8 heads per k
}

// ---- 1) value GEMM: (B*L,512) @ (512,512) -> value (B,H,L,DV) -------------
__global__ __launch_bounds__(256) void k_value(const float* __restrict__ inp,
                                               const float* __restrict__ Wv,
                                               float* __restrict__ value) {
  int wave = (blockIdx.x * blockDim.x + threadIdx.x) >> 5; // 0..2047
  int lane = threadIdx.x & 31;
  int mt = wave >> 5;  // 64 row tiles
  int nt = wave & 31;  // 32 col tiles
  int row0 = mt * 16, col0 = nt * 16;
  v8f acc = wmma_tile(inp, Dn, Wv, Hn * DVn, row0, col0, Dn, lane);
  int n = col0 + (lane & 15);
  int h = n >> 6, dv = n & 63;
#pragma unroll
  for (int v = 0; v < 8; ++v) {
    int m = row0 + v + ((lane >> 4) << 3);
    int b = m >> 7, l = m & 127;
    value[(((size_t)(b * Hn + h) * Ln) + l) * DVn + dv] = acc[v];
  }
}

// ---- 2) e_src/e_tgt : per (b,h,l) dot over DV ------------------------------
__global__ __launch_bounds__(256) void k_e(const float* __restrict__ value,
                                           const float* __restrict__ wsrc,
                                           const float* __restrict__ wtgt,
                                           float* __restrict__ e_src,
                                           float* __restrict__ e_tgt) {
  int id = blockIdx.x * blockDim.x + threadIdx.x; // 0..8191 = (b*H+h)*L + l
  int h = (id >> 7) & 7;
  const float* row = value + (size_t)id * DVn;
  const float* s = wsrc + h * DVn;
  const float* t = wtgt + h * DVn;
  float a = 0.0f, c = 0.0f;
#pragma unroll 8
  for (int d = 0; d < DVn; ++d) {
    float v = row[d];
    a = fmaf(v, s[d], a);
    c = fmaf(v, t[d], c);
  }
  e_src[id] = a;
  e_tgt[id] = c;
}

// ---- shared softmax/renorm for one 128-wide row (one wave) ----------------
// sc[jj] already has -INF where bad; adj4/bad4 per element; writes out[j].
__device__ __forceinline__ void softmax_row(const float sc[4], const int adj4[4],
                                            const bool bad4[4],
                                            float* __restrict__ out, int lane) {
  float m = fmaxf(fmaxf(sc[0], sc[1]), fmaxf(sc[2], sc[3]));
  m = wave_max(m);
  float p[4], lsum = 0.0f;
#pragma unroll
  for (int jj = 0; jj < 4; ++jj) {
    p[jj] = __expf(sc[jj] - m);
    lsum += p[jj];
  }
  float sum = wave_sum(lsum);
  float inv = 1.0f / sum;
  float q[4], asum = 0.0f;
#pragma unroll
  for (int jj = 0; jj < 4; ++jj) {
    q[jj] = p[jj] * inv * (float)adj4[jj];
    asum += fabsf(q[jj]);
  }
  asum = wave_sum(asum);
  float rinv = 1.0f / fmaxf(asum, 1e-12f);
#pragma unroll
  for (int jj = 0; jj < 4; ++jj) {
    float r = bad4[jj] ? 0.0f : q[jj] * rinv;
    out[lane + jj * 32] = r;
  }
}

// ---- 3) value-attention rows ----------------------------------------------
__global__ __launch_bounds__(256) void k_attn_value(
    const float* __restrict__ e_src, const float* __restrict__ e_tgt,
    const unsigned char* __restrict__ mask, const int* __restrict__ adj,
    float* __restrict__ attnV) {
  int g = (blockIdx.x * blockDim.x + threadIdx.x) >> 5; // (b*H+h)*L + i
  int lane = threadIdx.x & 31;
  int b = g >> 10, rem = g & 1023, h = rem >> 7, i = rem & 127;
  float et = e_tgt[g];
  const float* es = e_src + (g - i); // row base (b,h,0)
  const unsigned char* mrow = mask + ((size_t)(b * Ln + i)) * Ln;
  const int* arow = adj + ((size_t)g) * Ln;
  float sc[4];
  int a4[4];
  bool bad4[4];
#pragma unroll
  for (int jj = 0; jj < 4; ++jj) {
    int j = lane + jj * 32;
    float s = leaky(et + es[j]);
    int a = arow[j];
    bool bad = (mrow[j] != 0) || (a == 0);
    sc[jj] = bad ? -NEG_INF : s;
    a4[jj] = a;
    bad4[jj] = bad;
  }
  softmax_row(sc, a4, bad4, attnV + (size_t)g * Ln, lane);
}

// ---- 4) fused relation pass: stream relation_inp once ---------------------
// block = (b,i); 8 waves x 16 j-rows each; then per-head softmax.
__global__ __launch_bounds__(256) void k_rel(const float* __restrict__ rel,
                                             const float* __restrict__ wrc_g,
                                             const unsigned char* __restrict__ mask,
                                             const int* __restrict__ adj,
                                             float* __restrict__ attnR) {
  __shared__ float s_wrc[Dn * Hn]; // 16 KB
  __shared__ float s_sc[Hn * Ln];  // 4 KB
  int b = blockIdx.x >> 7, i = blockIdx.x & 127;
  for (int t = threadIdx.x; t < Dn * Hn; t += 256) s_wrc[t] = wrc_g[t];
  __syncthreads();
  int wave = threadIdx.x >> 5, lane = threadIdx.x & 31;
  for (int j = wave; j < Ln; j += 8) {
    const float* r = rel + (((size_t)(b * Ln + i) * Ln) + j) * Dn;
    float acc[8] = {0, 0, 0, 0, 0, 0, 0, 0};
#pragma unroll
    for (int c = 0; c < 4; ++c) {
      int k = c * 128 + lane * 4;
      if (j + 8 < Ln) __builtin_prefetch(r + 8 * Dn + k, 0, 0);
      float4 rv = *(const float4*)(r + k);
#pragma unroll
      for (int h = 0; h < 8; ++h) {
        float s = acc[h];
        s = fmaf(rv.x, s_wrc[(k + 0) * 8 + h], s);
        s = fmaf(rv.y, s_wrc[(k + 1) * 8 + h], s);
        s = fmaf(rv.z, s_wrc[(k + 2) * 8 + h], s);
        s = fmaf(rv.w, s_wrc[(k + 3) * 8 + h], s);
        acc[h] = s;
      }
    }
#pragma unroll
    for (int h = 0; h < 8; ++h) {
      float v = wave_sum(acc[h]);
      if (lane == 0) s_sc[h * Ln + j] = leaky(v);
    }
  }
  __syncthreads();
  int h = wave; // one wave per head row
  const unsigned char* mrow = mask + ((size_t)(b * Ln + i)) * Ln;
  size_t gi = ((size_t)(b * Hn + h) * Ln + i);
  const int* arow = adj + gi * Ln;
  float sc[4];
  int a4[4];
  bool bad4[4];
#pragma unroll
  for (int jj = 0; jj < 4; ++jj) {
    int j = lane + jj * 32;
    float s = s_sc[h * Ln + j];
    int a = arow[j];
    bool bad = (mrow[j] != 0) || (a == 0);
    sc[jj] = bad ? -NEG_INF : s;
    a4[jj] = a;
    bad4[jj] = bad;
  }
  softmax_row(sc, a4, bad4, attnR + gi * Ln, lane);
}

// ---- 5) out_v/out_r: per (b,h) 128x64x128 GEMMs into cat buffer -----------
__global__ __launch_bounds__(256) void k_av(const float* __restrict__ attnV,
                                            const float* __restrict__ attnR,
                                            const float* __restrict__ value,
                                            float* __restrict__ cat) {
  int g = (blockIdx.x * blockDim.x + threadIdx.x) >> 5; // 0..4095
  int lane = threadIdx.x & 31;
  int mat = g & 1;
  int t = g >> 1;
  int bh = t >> 5;          // 0..63
  int tile = t & 31;        // 8 x 4 tiles
  int mt = tile >> 2, nt = tile & 3;
  const float* A = (mat ? attnR : attnV) + (size_t)bh * Ln * Ln;
  const float* Bm = value + (size_t)bh * Ln * DVn;
  int row0 = mt * 16, col0 = nt * 16;
  v8f acc = wmma_tile(A, Ln, Bm, DVn, row0, col0, Ln, lane);
  int b = bh >> 3, h = bh & 7;
  int dv = col0 + (lane & 15);
#pragma unroll
  for (int v = 0; v < 8; ++v) {
    int i = row0 + v + ((lane >> 4) << 3);
    cat[((size_t)(b * Ln + i)) * (2 * Dn) + mat * Dn + h * DVn + dv] = acc[v];
  }
}

// ---- 6) final GEMM: (1024,1024) @ (1024,512) + bias + residual ------------
__global__ __launch_bounds__(256) void k_final(const float* __restrict__ cat,
                                               const float* __restrict__ Wf,
                                               const float* __restrict__ bf,
                                               const float* __restrict__ inp,
                                               float* __restrict__ out) {
  int wave = (blockIdx.x * blockDim.x + threadIdx.x) >> 5; // 0..2047
  int lane = threadIdx.x & 31;
  int mt = wave >> 5, nt = wave & 31;
  int row0 = mt * 16, col0 = nt * 16;
  v8f acc = wmma_tile(cat, 2 * Dn, Wf, Dn, row0, col0, 2 * Dn, lane);
  int n = col0 + (lane & 15);
  float bias = bf[n];
#pragma unroll
  for (int v = 0; v < 8; ++v) {
    int m = row0 + v + ((lane >> 4) << 3);
    out[(size_t)m * Dn + n] = acc[v] + bias + inp[(size_t)m * Dn + n];
  }
}

extern "C" void kernel_launch(void* const* d_in, const int* in_sizes, int n_in,
                              void* d_out, int out_size, void* d_ws,
                              size_t ws_size, hipStream_t stream) {
  const float* inp = (const float*)d_in[0];
  const float* rel = (const float*)d_in[1];
  const unsigned char* mask = (const unsigned char*)d_in[2];
  const int* adj = (const int*)d_in[3];
  const float* Wv = (const float*)d_in[7];
  const float* Wr = (const float*)d_in[8];
  const float* wsrc = (const float*)d_in[9];
  const float* wtgt = (const float*)d_in[10];
  const float* wrel = (const float*)d_in[11];
  const float* Wf = (const float*)d_in[12];
  const float* bfin = (const float*)d_in[13];

  float* ws = (float*)d_ws;
  float* wrc = ws;                      // 4096
  float* value = wrc + 4096;            // B*H*L*DV = 524288
  float* e_src = value + Bn * Hn * Ln * DVn;
  float* e_tgt = e_src + Bn * Hn * Ln;  // 8192 each
  float* cat = e_tgt + Bn * Hn * Ln;    // B*L*1024 = 1048576

  float* out_final = (float*)d_out;                     // (B,L,D)
  float* attnV = out_final + (size_t)Bn * Ln * Dn;      // (B,H,L,L)
  float* attnR = attnV + (size_t)Bn * Hn * Ln * Ln;     // (B,H,L,L)

  k_wrc<<<16, 256, 0, stream>>>(Wr, wrel, wrc);
  k_value<<<256, 256, 0, stream>>>(inp, Wv, value);
  k_e<<<32, 256, 0, stream>>>(value, wsrc, wtgt, e_src, e_tgt);
  k_attn_value<<<1024, 256, 0, stream>>>(e_src, e_tgt, mask, adj, attnV);
  k_rel<<<1024, 256, 0, stream>>>(rel, wrc, mask, adj, attnR);
  k_av<<<512, 256, 0, stream>>>(attnV, attnR, value, cat);
  k_final<<<256, 256, 0, stream>>>(cat, Wf, bfin, inp, out_final);
}